// MTL_binning_heads_48026324304413
// MI455X (gfx1250) — compile-verified
//
#include <hip/hip_runtime.h>

typedef float v2f __attribute__((ext_vector_type(2)));
typedef float v4f __attribute__((ext_vector_type(4)));
typedef float v8f __attribute__((ext_vector_type(8)));

#define B_BATCH 2048
#define NBINS   200
#define DDIM    128
#define GGENES  100
#define GPAD    112        // 7 * 16
#define LDS_STRIDE 132     // 128 + 4 pad: 132*4 = 528 B (16B-aligned rows, 4-bank rotation)

// LDS layout (Ws FIRST so hot-loop ds_load_b64 offsets fit the 16-bit immediate):
//   Ws   : [0            , 59136)   112*132 floats (g rows 100..111 zeroed)
//   bs   : [59136        , 59584)   112 floats
//   Xs   : [59584        , 127168)  128*132 floats
//
// One block = one bin x 128 batch rows. 8 waves, each computes a 16-row m-tile
// across seven 16-col g-tiles using V_WMMA_F32_16X16X4_F32, K=128 in 32 steps.
__global__ __launch_bounds__(256)
void mtl_binning_heads_wmma(const float* __restrict__ X,   // [B, NBINS, D]
                            const float* __restrict__ W,   // [NBINS, G, D]
                            const float* __restrict__ bias,// [NBINS, G]
                            float* __restrict__ out)       // [NBINS, B, G]
{
    extern __shared__ float smem[];
    float* Ws = smem;                                  // 112 * 132  (low LDS!)
    float* bs = smem + GPAD * LDS_STRIDE;              // 112
    float* Xs = bs + GPAD;                             // 128 * 132

    const int tid    = threadIdx.x;
    const int bin    = blockIdx.y;
    const int b_base = blockIdx.x * 128;

    // ---- Stage W tile [112 x 128] (rows = genes), zero-pad rows >= 100. ----
    {
        const float* wg = W + (size_t)bin * GGENES * DDIM;
        #pragma unroll
        for (int i = 0; i < 14; ++i) {                 // 3584 float4 / 256 threads
            int idx = tid + i * 256;
            int row = idx >> 5;
            int c4  = (idx & 31) << 2;
            v4f v = {0.f, 0.f, 0.f, 0.f};
            if (row < GGENES) v = *(const v4f*)(wg + row * DDIM + c4);
            *(v4f*)(Ws + row * LDS_STRIDE + c4) = v;
        }
    }
    // ---- Stage X tile [128 x 128] (rows = batch). Global row stride = NBINS*DDIM. ----
    {
        const float* xg = X + ((size_t)b_base * NBINS + bin) * DDIM;
        #pragma unroll
        for (int i = 0; i < 16; ++i) {                 // 4096 float4 / 256 threads
            int idx = tid + i * 256;
            int row = idx >> 5;
            int c4  = (idx & 31) << 2;
            v4f v = *(const v4f*)(xg + (size_t)row * (NBINS * DDIM) + c4);
            *(v4f*)(Xs + row * LDS_STRIDE + c4) = v;
        }
    }
    if (tid < GPAD) bs[tid] = (tid < GGENES) ? bias[bin * GGENES + tid] : 0.f;
    __syncthreads();

    const int lane = tid & 31;
    const int wave = tid >> 5;
    const int lo   = lane & 15;      // row within tile (A) / column (B, C/D)
    const int hi   = lane >> 4;      // lane-half: selects K pair (A/B) / M half (C/D)

    // ---- A fragments for this wave's m-tile, held in VGPRs across all g-tiles.
    // ISA layout (16x4 f32 A): lane holds M = lane%16, K = {2*hi, 2*hi+1} per step.
    v2f a[32];
    {
        const float* xrow = Xs + (wave * 16 + lo) * LDS_STRIDE + 2 * hi;
        #pragma unroll
        for (int s = 0; s < 32; ++s)
            a[s] = *(const v2f*)(xrow + 4 * s);
    }

    const int m_lane = b_base + wave * 16 + 8 * hi;    // C/D: VGPR r -> row m_lane + r

    for (int gt = 0; gt < 7; ++gt) {
        const int g = gt * 16 + lo;
        // B (4x16 f32): lane holds N = lane%16 = g, K = {2*hi, 2*hi+1} -> W[g][k..k+1]
        // Base address < 59 KB, per-step immediates 16*s < 512 B -> pure immediate
        // offsets in ds_load_b64, no per-load VALU address math.
        const float* wrow = Ws + g * LDS_STRIDE + 2 * hi;

        v8f acc = {};
        #pragma unroll
        for (int s = 0; s < 32; ++s) {
            v2f b2 = *(const v2f*)(wrow + 4 * s);
            acc = __builtin_amdgcn_wmma_f32_16x16x4_f32(
                      /*neg_a=*/false, a[s], /*neg_b=*/false, b2,
                      /*c_mod=*/(short)0, acc, /*reuse_a=*/false, /*reuse_b=*/false);
        }

        const float bb = bs[g];                        // zero-padded, safe for g >= 100
        if (g < GGENES) {
            float* o = out + ((size_t)bin * B_BATCH + m_lane) * GGENES + g;
            #pragma unroll
            for (int r = 0; r < 8; ++r)
                o[(size_t)r * GGENES] = acc[r] + bb;
        }
    }
}

extern "C" void kernel_launch(void* const* d_in, const int* in_sizes, int n_in,
                              void* d_out, int out_size, void* d_ws, size_t ws_size,
                              hipStream_t stream) {
    const float* X    = (const float*)d_in[0];  // [2048, 200, 128]
    const float* W    = (const float*)d_in[1];  // [200, 100, 128]
    const float* bias = (const float*)d_in[2];  // [200, 100]
    float* out        = (float*)d_out;          // [200, 2048, 100] flat

    dim3 grid(B_BATCH / 128, NBINS);
    size_t shmem = (size_t)(GPAD * LDS_STRIDE + GPAD + 128 * LDS_STRIDE) * sizeof(float);
    mtl_binning_heads_wmma<<<grid, 256, shmem, stream>>>(X, W, bias, out);
}